// JpegCompression_patch_17952963297637
// MI455X (gfx1250) — compile-verified
//
#include <hip/hip_runtime.h>

typedef float v2f __attribute__((ext_vector_type(2)));
typedef float v8f __attribute__((ext_vector_type(8)));

// JPEG luma quant table (row-major 8x8) and its compile-time reciprocals
__device__ __constant__ float c_YT[64] = {
  16,11,10,16,24,40,51,61,
  12,12,14,19,26,58,60,55,
  14,13,16,24,40,57,69,56,
  14,17,22,29,51,87,80,62,
  18,22,37,56,68,109,103,77,
  24,35,55,64,81,104,113,92,
  49,64,78,87,103,121,120,101,
  72,92,95,98,112,100,103,99 };

__device__ __constant__ float c_YTinv[64] = {
  1.0f/16,1.0f/11,1.0f/10,1.0f/16,1.0f/24,1.0f/40,1.0f/51,1.0f/61,
  1.0f/12,1.0f/12,1.0f/14,1.0f/19,1.0f/26,1.0f/58,1.0f/60,1.0f/55,
  1.0f/14,1.0f/13,1.0f/16,1.0f/24,1.0f/40,1.0f/57,1.0f/69,1.0f/56,
  1.0f/14,1.0f/17,1.0f/22,1.0f/29,1.0f/51,1.0f/87,1.0f/80,1.0f/62,
  1.0f/18,1.0f/22,1.0f/37,1.0f/56,1.0f/68,1.0f/109,1.0f/103,1.0f/77,
  1.0f/24,1.0f/35,1.0f/55,1.0f/64,1.0f/81,1.0f/104,1.0f/113,1.0f/92,
  1.0f/49,1.0f/64,1.0f/78,1.0f/87,1.0f/103,1.0f/121,1.0f/120,1.0f/101,
  1.0f/72,1.0f/92,1.0f/95,1.0f/98,1.0f/112,1.0f/100,1.0f/103,1.0f/99 };

// Orthonormal 8-point DCT-II matrix C8[u][x] = 0.5*alpha(u)*cos((2x+1)u*pi/16)
#define DA 0.35355339059327376f  // 0.5/sqrt(2)
#define DB 0.49039264020161522f  // 0.5*cos(pi/16)
#define DC 0.46193976625564338f  // 0.5*cos(2pi/16)
#define DD 0.41573480615127262f  // 0.5*cos(3pi/16)
#define DE 0.27778511650980111f  // 0.5*cos(5pi/16)
#define DF 0.19134171618254489f  // 0.5*cos(6pi/16)
#define DG 0.09754516100806413f  // 0.5*cos(7pi/16)
__device__ __constant__ float c_C8[64] = {
   DA,  DA,  DA,  DA,  DA,  DA,  DA,  DA,
   DB,  DD,  DE,  DG, -DG, -DE, -DD, -DB,
   DC,  DF, -DF, -DC, -DC, -DF,  DF,  DC,
   DD, -DG, -DB, -DE,  DE,  DB,  DG, -DD,
   DA, -DA, -DA,  DA,  DA, -DA, -DA,  DA,
   DE, -DB,  DG,  DD, -DD, -DG,  DB, -DE,
   DF, -DC,  DC, -DF, -DF,  DC, -DC,  DF,
   DG, -DE,  DD, -DB,  DB, -DD,  DE, -DG };

// 16x16x16 f32 matmul via 4 chained V_WMMA_F32_16X16X4_F32.
__device__ __forceinline__ v8f matmul16(v8f P, v8f B) {
  v8f acc = {};
#pragma unroll
  for (int s = 0; s < 4; ++s) {
    v2f a; a.x = P[2 * s]; a.y = P[2 * s + 1];
    v2f b; b.x = B[2 * s]; b.y = B[2 * s + 1];
    acc = __builtin_amdgcn_wmma_f32_16x16x4_f32(
        false, a, false, b, (short)0, acc, false, false);
  }
  return acc;
}

__global__ __launch_bounds__(256) void jpeg_dct_wmma_kernel(
    const float* __restrict__ x, float* __restrict__ out, int nch) {
  const int lane = threadIdx.x & 31;
  const int wave = threadIdx.x >> 5;
  const int h = lane >> 4;     // half of wave
  const int n = lane & 15;     // column (C/D/B layouts) / row m (A-op layout)

  // k-column held by A-op / B-op register a for this lane's half
  int kk[8];
#pragma unroll
  for (int a = 0; a < 8; ++a) kk[a] = 4 * (a >> 1) + 2 * h + (a & 1);

  // Constant B-operand fragments (once per wave, table-driven, no trig).
  // B1: rows = A^T ; B2: rows k = A^T[r(k)] ; B3: rows k = A[r(k)]
  v8f B1, B2, B3;
#pragma unroll
  for (int a = 0; a < 8; ++a) {
    int k = kk[a];
    int rk = 2 * (k >> 2) + (k & 1) + 8 * ((k >> 1) & 1);  // C->A perm
    B1[a] = ((n >> 3) == (k >> 3))  ? c_C8[(n & 7) * 8 + (k & 7)]  : 0.0f;
    B2[a] = ((n >> 3) == (rk >> 3)) ? c_C8[(n & 7) * 8 + (rk & 7)] : 0.0f;
    B3[a] = ((rk >> 3) == (n >> 3)) ? c_C8[(rk & 7) * 8 + (n & 7)] : 0.0f;
  }

  // Quant/dequant values for D^T in C layout: table Y_T[n&7][j], factor 0.02
  float qv[8], qinv[8];
#pragma unroll
  for (int j = 0; j < 8; ++j) {
    qv[j]   = c_YT[(n & 7) * 8 + j]    * 0.02f;
    qinv[j] = c_YTinv[(n & 7) * 8 + j] * 50.0f;   // no runtime division
  }

  // Precompute per-lane gather indices (edge-padded)
  const int sr = min(max(n - 1, 0), 13);
  int si[8];
#pragma unroll
  for (int a = 0; a < 8; ++a) si[a] = sr * 14 + min(max(kk[a] - 1, 0), 13);

  const int nwaves = gridDim.x * 8;
  const bool wr = (n >= 1 && n <= 14);

  for (int ch = blockIdx.x * 8 + wave; ch < nch; ch += nwaves) {
    const float* src = x + (size_t)ch * 196;
    // Prefetch next channel this wave will process
    if (ch + nwaves < nch)
      __builtin_prefetch(x + (size_t)(ch + nwaves) * 196 + si[0], 0, 0);

    // Load padded image directly in A-operand layout: element (m=n, k=kk[a]).
    float r[8];
#pragma unroll
    for (int a = 0; a < 8; ++a) r[a] = src[si[a]];

    // Channel min/max (padding only duplicates values).
    float mn = r[0], mx = r[0];
#pragma unroll
    for (int a = 1; a < 8; ++a) { mn = fminf(mn, r[a]); mx = fmaxf(mx, r[a]); }
#pragma unroll
    for (int off = 16; off >= 1; off >>= 1) {
      mn = fminf(mn, __shfl_xor(mn, off, 32));
      mx = fmaxf(mx, __shfl_xor(mx, off, 32));
    }
    const float rng = mx - mn + 1e-5f;
    // 1/rng via v_rcp_f32 + one Newton step (~0.5 ulp)
    float rc = __builtin_amdgcn_rcpf(rng);
    rc = rc * (2.0f - rng * rc);
    const float s255 = 255.0f * rc;

    v8f X;
#pragma unroll
    for (int a = 0; a < 8; ++a) X[a] = (r[a] - mn) * s255 - 128.0f;

    // Forward DCT:  T1 = X*A^T ;  T2 = (T1)^T*A^T = D^T
    v8f T1 = matmul16(X, B1);
    v8f T2 = matmul16(T1, B2);

    // Differentiable quantize/dequantize on D^T
    v8f Q;
#pragma unroll
    for (int j = 0; j < 8; ++j) {
      float d = T2[j] * qinv[j];
      float rn = rintf(d);           // v_rndne_f32
      float e = d - rn;
      Q[j] = (rn + e * e * e) * qv[j];
    }

    // Inverse DCT:  T3 = Q_d*A ;  T4 = (T3)^T*A = X2^T
    v8f T3 = matmul16(Q, B3);
    v8f T4 = matmul16(T3, B3);

    // Lane holds X2(r=n, c=j+8h). Crop [1,14], clip, rescale, store.
    if (wr) {
      float* dst = out + (size_t)ch * 196 + (size_t)(n - 1) * 14;
#pragma unroll
      for (int j = 0; j < 8; ++j) {
        int rr2 = j + 8 * h;
        if (rr2 >= 1 && rr2 <= 14) {
          float p = fminf(fmaxf(T4[j] + 128.0f, 0.0f), 255.0f);
          dst[rr2 - 1] = p * (1.0f / 255.0f) * rng + mn;
        }
      }
    }
  }
}

extern "C" void kernel_launch(void* const* d_in, const int* in_sizes, int n_in,
                              void* d_out, int out_size, void* d_ws, size_t ws_size,
                              hipStream_t stream) {
  (void)n_in; (void)d_ws; (void)ws_size; (void)out_size;
  const float* x = (const float*)d_in[0];
  float* out = (float*)d_out;
  int nch = in_sizes[0] / 196;             // 32*1024 channels of 14x14
  dim3 block(256);                         // 8 waves per block
  dim3 grid(1024);                         // 8192 waves, 4 channels per wave
  jpeg_dct_wmma_kernel<<<grid, block, 0, stream>>>(x, out, nch);
}